// decode_param_CNN_32384053412562
// MI455X (gfx1250) — compile-verified
//
#include <hip/hip_runtime.h>
#include <cstddef>

// Problem constants (from reference): N_B=4, N_T=16, N_Y=N_X=256
#define NB 4
#define NT 16
#define NP 65536                          // N_Y * N_X spatial sites per plane
#define IN_BSTR ((size_t)112 * NP)        // floats per batch in input (7*NT planes)

// Block partition of the fused kernel (256 threads/block, all exact — no tails)
#define BLKS_H      1024                  // NB*NP / 256          (heavy: ~448 B/thread)
#define BLKS_M      8192                  // NB*32*NP/4 / 256     (copy)
#define BLKS_KAPPA  4096                  // NB*16*NP/4 / 256     (softplus)
#define BLKS_TAU    4096                  // NB*16*NP/4 / 256     (softplus)
#define BLKS_TOTAL  (BLKS_H + BLKS_M + BLKS_KAPPA + BLKS_TAU)

// Native clang vector type so __builtin_nontemporal_load/store accepts it
// (HIP's float4 is a struct and is rejected by the builtin).
typedef __attribute__((ext_vector_type(4))) float v4f;

// softplus(10x)/10, numerically stable: max(z,0) + log1p(exp(-|z|)), z = 10x
__device__ __forceinline__ float softplus_b10(float x) {
    float z = 10.0f * x;
    return 0.1f * (fmaxf(z, 0.0f) + log1pf(__expf(-fabsf(z))));
}

// Elementwise region body: output flat layout == input slice flat layout
// (the reference reshapes for kappa/m/tau are pure reinterpretations).
// b128 NT streaming: 246 MB working set > 192 MB L2, every byte touched once.
template <bool SP>
__device__ __forceinline__ void ew_body(const float* __restrict__ in,
                                        float* __restrict__ out,
                                        int i,          // v4f index within region
                                        int n4_per_b,   // v4f's per batch
                                        int in_off)     // float offset of slice within batch
{
    int b = i / n4_per_b;
    int r = i - b * n4_per_b;
    const v4f* src = (const v4f*)(in + (size_t)b * IN_BSTR + (size_t)in_off);
    v4f v = __builtin_nontemporal_load(&src[r]);
    if (SP) {
        v.x = softplus_b10(v.x);
        v.y = softplus_b10(v.y);
        v.z = softplus_b10(v.z);
        v.w = softplus_b10(v.w);
    }
    __builtin_nontemporal_store(v, ((v4f*)out) + i);
}

// H region body: H[b,i,j,p,t] = (i==j)*softplus10(gamma[b,t,p]) + v_i*v_j.
// One thread per (b,p). Gather: for each t, consecutive lanes hit consecutive
// addresses -> each b32 NT load is one full 128B wave transaction (compiler
// emits s_clause'd loads off one 64-bit base + immediate offsets). Scatter:
// 16 contiguous t-values per (i,j) plane -> 4x b128 NT stores; a wave covers
// a contiguous 2KB span. The layout transpose is done purely via addressing.
__device__ __forceinline__ void h_body(const float* __restrict__ in,
                                       float* __restrict__ outH,
                                       int idx)        // b*NP + p
{
    int b = idx >> 16;          // idx / NP  (NP == 1<<16)
    int p = idx & (NP - 1);

    const float* base = in + (size_t)b * IN_BSTR + (size_t)p;

    float g[NT], vx[NT], vy[NT];
#pragma unroll
    for (int t = 0; t < NT; ++t) {
        vx[t] = __builtin_nontemporal_load(base + (size_t)(64 + t) * NP);
        vy[t] = __builtin_nontemporal_load(base + (size_t)(80 + t) * NP);
        g[t]  = softplus_b10(__builtin_nontemporal_load(base + (size_t)(48 + t) * NP));
    }

    float h00[NT], h01[NT], h11[NT];
#pragma unroll
    for (int t = 0; t < NT; ++t) {
        h00[t] = fmaf(vx[t], vx[t], g[t]);   // gamma + vx*vx
        h01[t] = vx[t] * vy[t];              // off-diagonal (symmetric)
        h11[t] = fmaf(vy[t], vy[t], g[t]);   // gamma + vy*vy
    }

    const size_t plane = (size_t)NP * NT;                 // elements per (b,i,j) plane
    float* o00 = outH + ((size_t)b * 4 + 0) * plane + (size_t)p * NT;
    float* o01 = o00 + plane;
    float* o10 = o01 + plane;
    float* o11 = o10 + plane;

#pragma unroll
    for (int t = 0; t < NT; t += 4) {
        v4f q00 = { h00[t], h00[t + 1], h00[t + 2], h00[t + 3] };
        v4f q01 = { h01[t], h01[t + 1], h01[t + 2], h01[t + 3] };
        v4f q11 = { h11[t], h11[t + 1], h11[t + 2], h11[t + 3] };
        __builtin_nontemporal_store(q00, (v4f*)(o00 + t));
        __builtin_nontemporal_store(q01, (v4f*)(o01 + t));
        __builtin_nontemporal_store(q01, (v4f*)(o10 + t));   // H symmetric
        __builtin_nontemporal_store(q11, (v4f*)(o11 + t));
    }
}

// ---------------------------------------------------------------------------
// Single fused kernel: one launch instead of four (graph capture serializes
// same-stream launches, and the 3 extra launch/tail
// boundaries are a real cost). Region selected by blockIdx.x range; every
// region is an exact multiple of 256 threads so no bounds checks. Heavy H
// blocks go first so long-running work starts earliest.
// ---------------------------------------------------------------------------
__global__ __launch_bounds__(256) void decode_fused_kernel(
    const float* __restrict__ in,
    float* __restrict__ out_kappa,
    float* __restrict__ out_m,
    float* __restrict__ out_H,
    float* __restrict__ out_tau)
{
    int blk = blockIdx.x;
    int tid = threadIdx.x;

    if (blk < BLKS_H) {
        // H: transposing gather + 2x2 symmetric outer product
        h_body(in, out_H, blk * 256 + tid);
    } else if (blk < BLKS_H + BLKS_M) {
        // m = params[:, 16:48] verbatim (two contiguous input blocks per batch)
        int i = (blk - BLKS_H) * 256 + tid;
        ew_body<false>(in, out_m, i, (32 * NP) / 4, 16 * NP);
    } else if (blk < BLKS_H + BLKS_M + BLKS_KAPPA) {
        // kappa = softplus10(params[:, 0:16]), layout-preserving
        int i = (blk - (BLKS_H + BLKS_M)) * 256 + tid;
        ew_body<true>(in, out_kappa, i, (16 * NP) / 4, 0);
    } else {
        // tau = softplus10(params[:, 96:112]), layout-preserving
        int i = (blk - (BLKS_H + BLKS_M + BLKS_KAPPA)) * 256 + tid;
        ew_body<true>(in, out_tau, i, (16 * NP) / 4, 96 * NP);
    }
}

extern "C" void kernel_launch(void* const* d_in, const int* in_sizes, int n_in,
                              void* d_out, int out_size, void* d_ws, size_t ws_size,
                              hipStream_t stream) {
    (void)in_sizes; (void)n_in; (void)out_size; (void)d_ws; (void)ws_size;

    const float* in = (const float*)d_in[0];
    float* out = (float*)d_out;

    // Output tuple flat layout: kappa | m | H | tau
    float* out_kappa = out;                                  // NB * 16*NP
    float* out_m     = out_kappa + (size_t)NB * 16 * NP;     // NB * 32*NP
    float* out_H     = out_m     + (size_t)NB * 32 * NP;     // NB * 64*NP
    float* out_tau   = out_H     + (size_t)NB * 64 * NP;     // NB * 16*NP

    decode_fused_kernel<<<BLKS_TOTAL, 256, 0, stream>>>(in, out_kappa, out_m, out_H, out_tau);
}